// WGCN_80771154969041
// MI455X (gfx1250) — compile-verified
//
#include <hip/hip_runtime.h>
#include <math.h>

// ---------------- CDNA5 WMMA f32 path ----------------
typedef float v2f __attribute__((ext_vector_type(2)));
typedef float v8f __attribute__((ext_vector_type(8)));
typedef unsigned int u32x4 __attribute__((ext_vector_type(4)));
typedef int i32x4 __attribute__((ext_vector_type(4)));
typedef int i32x8 __attribute__((ext_vector_type(8)));

#ifndef __has_builtin
#define __has_builtin(x) 0
#endif
#if __has_builtin(__builtin_amdgcn_tensor_load_to_lds) && \
    __has_builtin(__builtin_amdgcn_s_wait_tensorcnt)
#define HAVE_TDM 1
#else
#define HAVE_TDM 0
#endif

__device__ __forceinline__ v8f wmma4(v2f a, v2f b, v8f c) {
  // D = A(16x4) * B(4x16) + C(16x16), f32; (neg_a, A, neg_b, B, c_mod, C, reuse_a, reuse_b)
  return __builtin_amdgcn_wmma_f32_16x16x4_f32(false, a, false, b, (short)0, c,
                                               false, false);
}

#if HAVE_TDM
// TDM: DMA a K x 64 f32 tile (row stride ldb elements) from global into LDS.
// D# built per CDNA5 ISA 8.3/8.4: group0 = {count|flags, lds_addr, gaddr_lo,
// gaddr_hi|type=2}; group1 packs data_size=4B, tensor_dim0=64, tensor_dim1=K,
// tile_dim0=64, tile_dim1=K, tensor_dim0_stride=ldb.
__device__ __forceinline__ void tdm_load_b(const float* gsrc, float* ldst,
                                           int K, int ldb) {
  unsigned long long ga = (unsigned long long)(const void*)gsrc;
  unsigned lds = (unsigned)(unsigned long long)(void*)ldst;  // flat low 32 = LDS offset
  u32x4 g0 = {1u, lds, (unsigned)ga,
              (unsigned)((ga >> 32) & 0x1FFFFFFu) | (2u << 30)};
  i32x8 g1 = {(int)(2u << 16),               // wg_mask=0, data_size=2 (4B)
              (int)(64u << 16),              // tensor_dim0[15:0]=64
              (int)((unsigned)K << 16),      // tensor_dim0 hi=0, tensor_dim1[15:0]=K
              (int)(64u << 16),              // tensor_dim1 hi=0, tile_dim0=64
              (int)(K & 0xFFFF),             // tile_dim1=K, tile_dim2=0
              ldb,                           // tensor_dim0_stride lo
              0, 0};
  i32x4 z4 = {0, 0, 0, 0};
#if defined(__clang_major__) && (__clang_major__ >= 23)
  i32x8 z8 = {0, 0, 0, 0, 0, 0, 0, 0};
  __builtin_amdgcn_tensor_load_to_lds(g0, g1, z4, z4, z8, 0);
#else
  __builtin_amdgcn_tensor_load_to_lds(g0, g1, z4, z4, 0);
#endif
  __builtin_amdgcn_s_wait_tensorcnt(0);
}
#endif

// ---------------- C = A @ B : LDS-staged B, 16x64 strip per wave ----------------
// Block = 128 rows x 64 cols of C; B slice (K x 64, K<=256) staged in LDS once.
// Optional epilogue: per-column scale (colScale) and accOut += result.
__global__ void k_ab(const float* __restrict__ A, int lda,
                     const float* __restrict__ B, int ldb,
                     float* __restrict__ C, int ldc,
                     int M, int N, int K,
                     const float* __restrict__ colScale,
                     float* __restrict__ accOut) {
  __shared__ float Bs[256 * 64];                      // 64 KB of 320 KB WGP LDS
  const int lane = threadIdx.x & 31, wave = threadIdx.x >> 5;
  const int mblks = M >> 7;
  const int mblk = blockIdx.x % mblks;
  const int ngrp = blockIdx.x / mblks;
  const float* Bg = B + (size_t)ngrp * 64;
#if HAVE_TDM
  if (threadIdx.x < 32) tdm_load_b(Bg, Bs, K, ldb);   // one wave issues the DMA
#else
  for (int i = threadIdx.x; i < K * 64; i += 256)
    Bs[i] = Bg[(size_t)(i >> 6) * ldb + (i & 63)];
#endif
  __syncthreads();

  const int lmod = lane & 15, lhalf = lane >> 4;
  const int am = mblk * 128 + wave * 16 + lmod;
  const float* Arow = A + (size_t)am * lda;

  v8f c0, c1, c2, c3;
  for (int r = 0; r < 8; ++r) { c0[r] = 0.f; c1[r] = 0.f; c2[r] = 0.f; c3[r] = 0.f; }

  for (int k0 = 0; k0 < K; k0 += 4) {
    __builtin_prefetch(Arow + k0 + 32, 0, 1);         // global_prefetch_b8 ahead of A stream
    const int ka = k0 + 2 * lhalf;
    v2f a; a[0] = Arow[ka]; a[1] = Arow[ka + 1];
    const float* B0 = Bs + ka * 64;
    const float* B1 = Bs + (ka + 1) * 64;
    v2f b;
    b[0] = B0[lmod];      b[1] = B1[lmod];      c0 = wmma4(a, b, c0);
    b[0] = B0[lmod + 16]; b[1] = B1[lmod + 16]; c1 = wmma4(a, b, c1);
    b[0] = B0[lmod + 32]; b[1] = B1[lmod + 32]; c2 = wmma4(a, b, c2);
    b[0] = B0[lmod + 48]; b[1] = B1[lmod + 48]; c3 = wmma4(a, b, c3);
  }

#pragma unroll
  for (int t = 0; t < 4; ++t) {
    v8f cc = (t == 0) ? c0 : (t == 1) ? c1 : (t == 2) ? c2 : c3;
    const int col = ngrp * 64 + t * 16 + lmod;
    const float sc = colScale ? colScale[col] : 1.0f;
    for (int r = 0; r < 8; ++r) {
      const int row = mblk * 128 + wave * 16 + r + 8 * lhalf;
      const size_t off = (size_t)row * ldc + col;
      const float v = cc[r] * sc;
      C[off] = v;
      if (accOut) accOut[off] += v;                   // wave-exclusive tile: deterministic
    }
  }
}

// -------- Cpart[chunk] = A^T B over a K-chunk (deterministic K-split), 16x64/wave --------
__global__ void k_atb(const float* __restrict__ A, int lda,
                      const float* __restrict__ B, int ldb,
                      float* __restrict__ Cpart,
                      int M, int N, int kChunk) {
  const int lane = threadIdx.x & 31, wave = threadIdx.x >> 5;
  const int ngt = N >> 6;
  const int wt = blockIdx.x * 8 + wave;
  if (wt >= (M >> 4) * ngt) return;
  const int tm = wt / ngt, ng = wt % ngt;
  const int lmod = lane & 15, lhalf = lane >> 4;
  const int amCol = tm * 16 + lmod;                   // column of A == row of A^T
  const int bcol = ng * 64 + lmod;
  const int kBase = blockIdx.y * kChunk;

  v8f c0, c1, c2, c3;
  for (int r = 0; r < 8; ++r) { c0[r] = 0.f; c1[r] = 0.f; c2[r] = 0.f; c3[r] = 0.f; }

  for (int kk = 0; kk < kChunk; kk += 4) {
    const int ka = kBase + kk + 2 * lhalf;
    v2f a;
    a[0] = A[(size_t)ka * lda + amCol];
    a[1] = A[(size_t)(ka + 1) * lda + amCol];
    const float* Br0 = B + (size_t)ka * ldb;
    const float* Br1 = B + (size_t)(ka + 1) * ldb;
    v2f b;
    b[0] = Br0[bcol];      b[1] = Br1[bcol];      c0 = wmma4(a, b, c0);
    b[0] = Br0[bcol + 16]; b[1] = Br1[bcol + 16]; c1 = wmma4(a, b, c1);
    b[0] = Br0[bcol + 32]; b[1] = Br1[bcol + 32]; c2 = wmma4(a, b, c2);
    b[0] = Br0[bcol + 48]; b[1] = Br1[bcol + 48]; c3 = wmma4(a, b, c3);
  }

  float* Cc = Cpart + (size_t)blockIdx.y * M * N;
#pragma unroll
  for (int t = 0; t < 4; ++t) {
    v8f cc = (t == 0) ? c0 : (t == 1) ? c1 : (t == 2) ? c2 : c3;
    const int col = ng * 64 + t * 16 + lmod;
    for (int r = 0; r < 8; ++r) {
      const int row = tm * 16 + r + 8 * lhalf;
      Cc[(size_t)row * N + col] = cc[r];
    }
  }
}

__global__ void k_reduce_chunks(const float* __restrict__ part, float* __restrict__ outp,
                                size_t n, int nChunks) {
  size_t i  = (size_t)blockIdx.x * blockDim.x + threadIdx.x;
  size_t st = (size_t)gridDim.x * blockDim.x;
  for (; i < n; i += st) {
    float s = 0.0f;
    for (int cc = 0; cc < nChunks; ++cc) s += part[(size_t)cc * n + i];  // fixed order
    outp[i] = s;
  }
}

// ---------------- elementwise helpers ----------------
__global__ void k_zero(float* p, size_t n) {
  size_t i  = (size_t)blockIdx.x * blockDim.x + threadIdx.x;
  size_t st = (size_t)gridDim.x * blockDim.x;
  for (; i < n; i += st) p[i] = 0.0f;
}

__global__ void k_elem_log(float* p, size_t n) {
  size_t i  = (size_t)blockIdx.x * blockDim.x + threadIdx.x;
  size_t st = (size_t)gridDim.x * blockDim.x;
  for (; i < n; i += st) p[i] = logf(p[i]);
}

__global__ void k_scale_cols(float* p, const float* __restrict__ S, int nrows) {
  size_t n  = (size_t)nrows * 64;
  size_t i  = (size_t)blockIdx.x * blockDim.x + threadIdx.x;
  size_t st = (size_t)gridDim.x * blockDim.x;
  for (; i < n; i += st) p[i] *= 0.5f * S[i & 63];
}

__global__ void k_transpose(const float* __restrict__ V, float* __restrict__ Bt) {
  int i = blockIdx.x * blockDim.x + threadIdx.x;     // 64 blocks * 256 = 16384
  if (i >= 256 * 64) return;
  int j = i >> 6, k = i & 63;
  Bt[k * 256 + j] = V[j * 64 + k];
}

__global__ void k_init_B(float* B) {
  int i = blockIdx.x * blockDim.x + threadIdx.x;     // 16384
  if (i >= 256 * 64) return;
  int r = i >> 6, cc = i & 63;
  unsigned h = (unsigned)r * 1103515245u + (unsigned)cc * 12345u + 2654435761u;
  h ^= h >> 13; h *= 2246822519u; h ^= h >> 16;
  float noise = ((h & 0xFFFF) * (1.0f / 65536.0f)) - 0.5f;
  B[i] = (r == cc ? 1.0f : 0.0f) + 0.01f * noise;
}

// ---------------- dis = normalize_rows(exp(dis)) : one row per wave ----------------
__global__ void k_exp_norm(float* __restrict__ dis, int nrows) {
  const int lane = threadIdx.x & 31, wave = threadIdx.x >> 5;
  const int row = blockIdx.x * 8 + wave;
  if (row >= nrows) return;
  float* p = dis + (size_t)row * 64;
  float e0 = expf(p[lane]);
  float e1 = expf(p[lane + 32]);
  float s = e0 + e1;
  for (int m = 16; m >= 1; m >>= 1) s += __shfl_xor(s, m, 32);
  const float sc = (s > 0.0f) ? (1.0f / s) : 0.0f;
  p[lane]      = e0 * sc;
  p[lane + 32] = e1 * sc;
}

// ---------------- wave32 bitonic sort of 64 elements (2 per lane) ----------------
__device__ __forceinline__ void bitonic64_val(float& a0, float& a1, int lane) {
  for (int k = 2; k <= 64; k <<= 1) {
    for (int j = k >> 1; j >= 1; j >>= 1) {
      if (j == 32) {                      // in-lane exchange (only at k==64, ascending)
        float lo = fminf(a0, a1), hi = fmaxf(a0, a1);
        a0 = lo; a1 = hi;
      } else {
        {
          float b = __shfl_xor(a0, j, 32);
          bool asc   = ((lane & k) == 0);
          bool lower = ((lane & j) == 0);
          a0 = (lower == asc) ? fminf(a0, b) : fmaxf(a0, b);
        }
        {
          float b = __shfl_xor(a1, j, 32);
          bool asc   = (((lane + 32) & k) == 0);
          bool lower = ((lane & j) == 0);
          a1 = (lower == asc) ? fminf(a1, b) : fmaxf(a1, b);
        }
      }
    }
  }
}

__device__ __forceinline__ void bitonic64_pair(float& a0, float& a1, int& p0, int& p1,
                                               int lane) {
  for (int k = 2; k <= 64; k <<= 1) {
    for (int j = k >> 1; j >= 1; j >>= 1) {
      if (j == 32) {
        if (a0 > a1) { float tv = a0; a0 = a1; a1 = tv; int tp = p0; p0 = p1; p1 = tp; }
      } else {
        {
          float bv = __shfl_xor(a0, j, 32); int bp = __shfl_xor(p0, j, 32);
          bool asc = ((lane & k) == 0), lower = ((lane & j) == 0);
          bool keep = (lower == asc) ? (a0 <= bv) : (a0 >= bv);
          if (!keep) { a0 = bv; p0 = bp; }
        }
        {
          float bv = __shfl_xor(a1, j, 32); int bp = __shfl_xor(p1, j, 32);
          bool asc = (((lane + 32) & k) == 0), lower = ((lane & j) == 0);
          bool keep = (lower == asc) ? (a1 <= bv) : (a1 >= bv);
          if (!keep) { a1 = bv; p1 = bp; }
        }
      }
    }
  }
}

// ---------------- sliced-Wasserstein barycenter: one wave per output row ----------------
template <int DEG>
__global__ void k_bary(const float* __restrict__ dis, const int* __restrict__ idx,
                       const int* __restrict__ adj, float* __restrict__ outd, int nrows) {
  __shared__ float smem[8][64];
  const int lane = threadIdx.x & 31, wave = threadIdx.x >> 5;
  const int row = blockIdx.x * 8 + wave;
  if (row >= nrows) return;

  float t0 = 0.f, t1 = 0.f;   // target (rank order)
  float m0 = 0.f, m1 = 0.f;   // bary   (position order)
  for (int d = 0; d < DEG; ++d) {
    const int nb = adj[(size_t)row * DEG + d];
    if (d + 1 < DEG) {        // hide next gather behind the sort (global_prefetch_b8)
      const int nb2 = adj[(size_t)row * DEG + d + 1];
      __builtin_prefetch(dis + (size_t)nb2 * 64 + lane, 0, 1);
    }
    const float* src = dis + (size_t)nb * 64;
    float v0 = src[lane], v1 = src[lane + 32];
    m0 += v0; m1 += v1;
    bitonic64_val(v0, v1, lane);
    t0 += v0; t1 += v1;
  }
  const float inv = 1.0f / (float)DEG;
  t0 *= inv; t1 *= inv; m0 *= inv; m1 *= inv;

  for (int it = 0; it < 5; ++it) {
    float v0 = m0, v1 = m1;
    int p0 = lane, p1 = lane + 32;
    bitonic64_pair(v0, v1, p0, p1, lane);
    // bary[pos] -= lr * (sorted - target[rank]); scatter via wave-private LDS strip
    smem[wave][p0] = v0 - 0.1f * (v0 - t0);
    smem[wave][p1] = v1 - 0.1f * (v1 - t1);
    m0 = smem[wave][lane];    // DS ops of one wave are in-order (DScnt)
    m1 = smem[wave][lane + 32];
  }
  float* dst = outd + (size_t)idx[row] * 64;
  dst[lane]      = m0;
  dst[lane + 32] = m1;
}

// ---------------- 64x64 SPD: upper Cholesky (C = R^T R) then Rinv = R^-1 ----------------
__global__ void k_chol_rinv(const float* __restrict__ Cin, float* __restrict__ Rinv) {
  __shared__ float R[64][64];
  const int t = threadIdx.x;                // 64 threads
  for (int i = 0; i < 64; ++i) R[i][t] = Cin[i * 64 + t];
  __syncthreads();
  for (int jj = 0; jj < 64; ++jj) {
    if (t == jj) R[jj][jj] = sqrtf(fmaxf(R[jj][jj], 1e-20f));
    __syncthreads();
    const float dinv = 1.0f / R[jj][jj];
    if (t > jj) R[jj][t] *= dinv;
    __syncthreads();
    for (int i = jj + 1; i < 64; ++i)
      if (t >= i) R[i][t] -= R[jj][i] * R[jj][t];
    __syncthreads();
  }
  float xcol[64];
  for (int i = 0; i < 64; ++i) xcol[i] = 0.0f;
  xcol[t] = 1.0f / R[t][t];
  for (int i = t - 1; i >= 0; --i) {
    float s = 0.0f;
    for (int k2 = i + 1; k2 <= t; ++k2) s += R[i][k2] * xcol[k2];
    xcol[i] = -s / R[i][i];
  }
  for (int i = 0; i < 64; ++i) Rinv[i * 64 + t] = (i <= t) ? xcol[i] : 0.0f;
}

// ---------------- singular values from converged subspace ----------------
__global__ void k_S(const float* __restrict__ V, const float* __restrict__ W,
                    float* __restrict__ S, float* __restrict__ iS) {
  const int k = threadIdx.x;                // 64 threads
  float s = 0.0f;
  for (int j = 0; j < 256; ++j) s += V[j * 64 + k] * W[j * 64 + k];
  const float sv = sqrtf(fmaxf(s, 0.0f));
  S[k]  = sv;
  iS[k] = (sv > 1e-20f) ? (1.0f / sv) : 0.0f;
}

// ==================================================================================
extern "C" void kernel_launch(void* const* d_in, const int* in_sizes, int n_in,
                              void* d_out, int out_size, void* d_ws, size_t ws_size,
                              hipStream_t stream) {
  (void)in_sizes; (void)n_in; (void)out_size; (void)ws_size;
  const float* x    = (const float*)d_in[0];
  const int* idx8   = (const int*)d_in[1];
  const int* adj8   = (const int*)d_in[2];
  const int* idx16  = (const int*)d_in[3];
  const int* adj16  = (const int*)d_in[4];
  const int* idx32  = (const int*)d_in[5];
  const int* adj32  = (const int*)d_in[6];
  float* out = (float*)d_out;

  const int N = 98304, D = 256, Cc = 64;
  float* ws = (float*)d_ws;
  size_t o = 0;
  float* dis  = ws + o; o += (size_t)N * Cc;
  float* tmp  = ws + o; o += (size_t)N * Cc;
  float* acc  = ws + o; o += (size_t)N * Cc;
  float* G    = ws + o; o += (size_t)D * D;
  float* V    = ws + o; o += (size_t)D * Cc;
  float* Bn   = ws + o; o += (size_t)D * Cc;
  float* W    = ws + o; o += (size_t)D * Cc;
  float* C64  = ws + o; o += (size_t)Cc * Cc;
  float* Rinv = ws + o; o += (size_t)Cc * Cc;
  float* S    = ws + o; o += Cc;
  float* iS   = ws + o; o += Cc;
  float* Bt   = ws + o; o += (size_t)Cc * D;
  float* part = ws + o; o += (size_t)48 * D * D;   // K-split partials (max use)

  dim3 blk(256);

  // ---- G = x^T x (48 deterministic K-chunks, ordered reduce) ----
  k_atb<<<dim3(8, 48), blk, 0, stream>>>(x, D, x, D, part, D, D, 2048);
  k_reduce_chunks<<<256, blk, 0, stream>>>(part, G, (size_t)D * D, 48);

  // ---- subspace iteration for top-64 right singular vectors ----
  k_init_B<<<64, blk, 0, stream>>>(V);
  for (int t = 0; t < 8; ++t) {
    k_ab<<<2, blk, 0, stream>>>(G, D, V, Cc, Bn, Cc, D, Cc, D, nullptr, nullptr);
    k_atb<<<dim3(1, 1), blk, 0, stream>>>(Bn, Cc, Bn, Cc, part, Cc, Cc, D);
    k_reduce_chunks<<<16, blk, 0, stream>>>(part, C64, (size_t)Cc * Cc, 1);
    k_chol_rinv<<<1, 64, 0, stream>>>(C64, Rinv);
    k_ab<<<2, blk, 0, stream>>>(Bn, Cc, Rinv, Cc, V, Cc, D, Cc, Cc, nullptr, nullptr);
  }
  // singular values: W = G V ; S_k = sqrt(v_k^T G v_k)
  k_ab<<<2, blk, 0, stream>>>(G, D, V, Cc, W, Cc, D, Cc, D, nullptr, nullptr);
  k_S<<<1, 64, 0, stream>>>(V, W, S, iS);

  // ---- U = x V S^-1  ->  dis ----
  k_ab<<<768, blk, 0, stream>>>(x, D, V, Cc, dis, Cc, N, Cc, D, iS, nullptr);
  k_zero<<<4096, blk, 0, stream>>>(acc, (size_t)N * Cc);

  float* cur = dis; float* alt = tmp;
  for (int layer = 0; layer < 2; ++layer) {
    k_exp_norm<<<12288, blk, 0, stream>>>(cur, N);
    for (int hop = 0; hop < 2; ++hop) {
      k_bary<8><<<4096, blk, 0, stream>>>(cur, idx8, adj8, alt, 32768);
      k_bary<16><<<4096, blk, 0, stream>>>(cur, idx16, adj16, alt, 32768);
      k_bary<32><<<4096, blk, 0, stream>>>(cur, idx32, adj32, alt, 32768);
      float* sw = cur; cur = alt; alt = sw;     // idx groups tile [0,N) exactly
    }
    k_elem_log<<<8192, blk, 0, stream>>>(cur, (size_t)N * Cc);
    // CholeskyQR: C = cur^T cur (96 chunks), R^-1, cur <- cur R^-1 ; acc += cur
    k_atb<<<dim3(1, 96), blk, 0, stream>>>(cur, Cc, cur, Cc, part, Cc, Cc, 1024);
    k_reduce_chunks<<<16, blk, 0, stream>>>(part, C64, (size_t)Cc * Cc, 96);
    k_chol_rinv<<<1, 64, 0, stream>>>(C64, Rinv);
    k_ab<<<768, blk, 0, stream>>>(cur, Cc, Rinv, Cc, alt, Cc, N, Cc, Cc, nullptr, acc);
    { float* sw = cur; cur = alt; alt = sw; }
  }

  // ---- out = (acc * 0.5 * S) @ base^T ----
  k_scale_cols<<<8192, blk, 0, stream>>>(acc, S, N);
  k_transpose<<<64, blk, 0, stream>>>(V, Bt);
  k_ab<<<3072, blk, 0, stream>>>(acc, Cc, Bt, D, out, D, N, D, Cc, nullptr, nullptr);
}